// SoftAttention_66614942761497
// MI455X (gfx1250) — compile-verified
//
#include <hip/hip_runtime.h>
#include <math.h>
#include <stdint.h>

// ---------------------------------------------------------------------------
// Additive (Bahdanau) attention for MI455X / gfx1250.
//   B=4, Q=64, S=1024, H=512, TEMPERATURE=1
// d_out = [context (4*64*512 f32) | weights (4*64*1024 f32)]
// d_ws  = [qp (256*512 f32) | kp (4096*512 f32)]  ~8.5 MB
//
// GEMMs run on V_WMMA_F32_16X16X4_F32 with K-panels staged into LDS by the
// Tensor Data Mover (tensor_load_to_lds, TENSORcnt), double buffered.
// The B*Q*S*H tanh tensor (dominant cost, 134M tanh) runs on v_tanh_f32.
// ---------------------------------------------------------------------------

#define Bc 4
#define Qc 64
#define Sc 1024
#define Hc 512

#define KB 32            // K-chunk (elements) staged per TDM transfer
#define APAD_STRIDE 33   // A panel LDS row stride (32 + 1 pad dword via TDM pad)

typedef __attribute__((ext_vector_type(2))) float v2f;
typedef __attribute__((ext_vector_type(8))) float v8f;
typedef __attribute__((ext_vector_type(4))) unsigned int v4u;
typedef __attribute__((ext_vector_type(4))) int v4i;
typedef __attribute__((ext_vector_type(8))) int v8i;

#if defined(__has_builtin)
#if __has_builtin(__builtin_amdgcn_tensor_load_to_lds)
#define HAVE_TDM 1
#endif
#endif
#ifndef HAVE_TDM
#define HAVE_TDM 0
#endif

// gfx1250 hardware tanh (TRANS unit, co-executes with VALU FMA)
__device__ __forceinline__ float fast_tanh(float x) {
#if __has_builtin(__builtin_amdgcn_tanhf)
  return __builtin_amdgcn_tanhf(x);
#elif __has_builtin(__builtin_amdgcn_tanh_f32)
  return __builtin_amdgcn_tanh_f32(x);
#else
  float r;
  asm volatile("v_tanh_f32 %0, %1\n\tv_nop\n\tv_nop" : "=v"(r) : "v"(x));
  return r;
#endif
}

#if HAVE_TDM
// ---------------------------------------------------------------------------
// Issue one TDM 2D tile load: global (tile_h rows x tile_w dwords, row stride
// row_stride dwords) -> LDS at lds_off. Optional LDS padding of 1 dword every
// (1<<(pad_code+1)) dwords. D# packing per CDNA5 ISA 8.3/8.4.
// ---------------------------------------------------------------------------
__device__ __forceinline__ void tdm_load_2d(unsigned lds_off, const float* gptr,
                                            unsigned tile_w, unsigned tile_h,
                                            unsigned long long row_stride,
                                            int pad_enable, unsigned pad_code) {
  const unsigned long long ga = (unsigned long long)(uintptr_t)gptr;
  v4u g0;
  g0.x = 0x1u;                                            // count=1, user mode
  g0.y = lds_off;                                         // lds_addr (bytes)
  g0.z = (unsigned)(ga & 0xffffffffu);                    // global_addr[31:0]
  g0.w = (unsigned)((ga >> 32) & 0x01ffffffu) | (2u << 30); // addr[56:32]|type=2

  v8i g1;
  unsigned w0 = (2u << 16);                               // data_size = 4B
  if (pad_enable) w0 |= (1u << 20) | (pad_code << 22);    // pad_amount=0 (1 dw)
  g1[0] = (int)w0;
  g1[1] = (int)((tile_w & 0xffffu) << 16);                // tensor_dim0 lo16
  g1[2] = (int)(((tile_w >> 16) & 0xffffu) | ((tile_h & 0xffffu) << 16)); // td0 hi | td1 lo
  g1[3] = (int)(((tile_h >> 16) & 0xffffu) | ((tile_w & 0xffffu) << 16)); // td1 hi | tile_dim0
  g1[4] = (int)(tile_h & 0xffffu);                        // tile_dim1 (tile_dim2=0)
  g1[5] = (int)(row_stride & 0xffffffffu);                // tensor_dim0_stride lo32
  g1[6] = (int)((row_stride >> 32) & 0xffffu);            // stride hi16 (dim1_stride=0)
  g1[7] = 0;
  const v4i gz = {0, 0, 0, 0};
#if __clang_major__ >= 23
  const v8i gz8 = {0, 0, 0, 0, 0, 0, 0, 0};
  __builtin_amdgcn_tensor_load_to_lds(g0, g1, gz, gz, gz8, 0);
#else
  __builtin_amdgcn_tensor_load_to_lds(g0, g1, gz, gz, 0);
#endif
}
#endif  // HAVE_TDM

// ---------------------------------------------------------------------------
// WMMA f32 GEMM:  C[M,N] = A[M,K] * B[K,N] (+ bias[N]).
// Block = 128 thr (4 waves), block tile 64x64; wave w owns rows w*16..w*16+15.
// K loop in KB=32 chunks: A panel (64x32, LDS stride 33) and B panel (32x64)
// staged by TDM (wave 0 issues; s_wait_tensorcnt + barrier), double buffered.
// M must be a multiple of 64 (true for all call sites) -> no EXEC divergence.
// ---------------------------------------------------------------------------
__global__ void wmma_gemm_bias(const float* __restrict__ A,
                               const float* __restrict__ Bm,
                               const float* __restrict__ bias,
                               float* __restrict__ C,
                               int M, int N, int K,
                               long long batchA, long long batchB, long long batchC) {
  __shared__ float sA[2][64 * APAD_STRIDE];
  __shared__ float sB[2][KB * 64];

  const int tid  = threadIdx.x;
  const int lane = tid & 31;
  const int wid  = tid >> 5;
  const int half = lane >> 4;   // 0: lanes 0-15, 1: lanes 16-31
  const int l16  = lane & 15;
  const int half2 = half * 2;

  const int m0 = blockIdx.y * 64;           // block row base
  const int n0 = blockIdx.x * 64;           // block col base

  const long long bz = blockIdx.z;
  A  += bz * batchA;
  Bm += bz * batchB;
  C  += bz * batchC;

  const float* Abase = A + (size_t)m0 * K;  // 64 x K panel source
  const float* Bbase = Bm + n0;             // K x 64 panel source

  const int nch = K / KB;

#if HAVE_TDM
  if (wid == 0) {
    tdm_load_2d((unsigned)(uintptr_t)&sA[0][0], Abase, KB, 64,
                (unsigned long long)K, 1, 4u /* pad every 32 dwords */);
    tdm_load_2d((unsigned)(uintptr_t)&sB[0][0], Bbase, 64, KB,
                (unsigned long long)N, 0, 0u);
  }
#else
  {  // cooperative fallback staging for chunk 0
    for (int i = tid; i < 64 * KB; i += 128) {
      const int r = i / KB, c = i % KB;
      sA[0][r * APAD_STRIDE + c] = Abase[(size_t)r * K + c];
    }
    for (int i = tid; i < KB * 64; i += 128) {
      const int r = i / 64, c = i % 64;
      sB[0][r * 64 + c] = Bbase[(size_t)r * N + c];
    }
  }
#endif

  v8f acc[4] = {};
  const int aoff = (wid * 16 + l16) * APAD_STRIDE;

  for (int ch = 0; ch < nch; ++ch) {
    const int cur = ch & 1;
#if HAVE_TDM
    if (wid == 0) __builtin_amdgcn_s_wait_tensorcnt(0);  // chunk ch landed
#endif
    __syncthreads();  // (a) LDS data visible, (b) prev buffer fully consumed

    if (ch + 1 < nch) {
      const int nxt = cur ^ 1;
      const int kb  = (ch + 1) * KB;
#if HAVE_TDM
      if (wid == 0) {
        tdm_load_2d((unsigned)(uintptr_t)&sA[nxt][0], Abase + kb, KB, 64,
                    (unsigned long long)K, 1, 4u);
        tdm_load_2d((unsigned)(uintptr_t)&sB[nxt][0], Bbase + (size_t)kb * N,
                    64, KB, (unsigned long long)N, 0, 0u);
      }
#else
      for (int i = tid; i < 64 * KB; i += 128) {
        const int r = i / KB, c = i % KB;
        sA[nxt][r * APAD_STRIDE + c] = Abase[(size_t)r * K + kb + c];
      }
      for (int i = tid; i < KB * 64; i += 128) {
        const int r = i / 64, c = i % 64;
        sB[nxt][r * 64 + c] = Bbase[(size_t)(kb + r) * N + c];
      }
#endif
    }

    // ---- compute chunk: 8 k-steps x 4 WMMAs, operands from LDS ----
    const float* Al = &sA[cur][0];
    const float* Bl = &sB[cur][0];
#pragma unroll
    for (int ks = 0; ks < KB; ks += 4) {
      const int kk = ks + half2;
      v2f av;
      av.x = Al[aoff + kk];
      av.y = Al[aoff + kk + 1];
      const float* b0 = Bl + kk * 64 + l16;
#pragma unroll
      for (int t = 0; t < 4; ++t) {
        v2f bv;
        bv.x = b0[t * 16];
        bv.y = b0[t * 16 + 64];
        acc[t] = __builtin_amdgcn_wmma_f32_16x16x4_f32(
            false, av, false, bv, (short)0, acc[t], false, false);
      }
    }
  }

  // ---- store C tile (+bias). C layout: VGPR i = row half*8+i, col l16 ----
  const int mrow = m0 + wid * 16 + half * 8;
#pragma unroll
  for (int t = 0; t < 4; ++t) {
    const int col = n0 + t * 16 + l16;
    const float bb = bias ? bias[col] : 0.0f;
#pragma unroll
    for (int i = 0; i < 8; ++i) {
      C[(size_t)(mrow + i) * N + col] = acc[t][i] + bb;
    }
  }
}

// ---------------------------------------------------------------------------
// Fused scores + softmax. One workgroup (256 thr = 8 waves) per (b,q).
// score[s] = we . tanh(qp[b,q,:] + kp[b,s,:]) + be -> softmax over S -> wout.
// Lanes split H=512 as 4 x float4; qp/we fragments live in registers; kp rows
// stream coalesced (L2-resident) with a 4-row-ahead prefetch.
// ---------------------------------------------------------------------------
__global__ void attn_scores_softmax(const float* __restrict__ qp,   // [B*Q, H]
                                    const float* __restrict__ kp,   // [B*S, H]
                                    const float* __restrict__ we,   // [H]
                                    const float* __restrict__ be,   // scalar
                                    float* __restrict__ wout) {     // [B*Q, S]
  const int bq   = blockIdx.x;        // 0..255
  const int b    = bq >> 6;
  const int tid  = threadIdx.x;       // 0..255
  const int lane = tid & 31;
  const int wid  = tid >> 5;          // 0..7

  __shared__ float sc[Sc];
  __shared__ float red[256];

  const float* qrow = qp + (size_t)bq * Hc;
  float4 qv[4], wv[4];
#pragma unroll
  for (int i = 0; i < 4; ++i) {
    const int h = i * 128 + lane * 4;
    qv[i] = *(const float4*)(qrow + h);
    wv[i] = *(const float4*)(we + h);
  }
  const float bias = be[0];

  const float* kbase = kp + (size_t)b * Sc * Hc;
  const int s_begin = wid * 128;
  for (int s = s_begin; s < s_begin + 128; ++s) {
    const float* krow = kbase + (size_t)s * Hc;
    __builtin_prefetch(krow + 4 * Hc + lane * 16, 0, 0);  // global_prefetch_b8
    float part = 0.0f;
#pragma unroll
    for (int i = 0; i < 4; ++i) {
      const float4 kv = *(const float4*)(krow + i * 128 + lane * 4);
      part += wv[i].x * fast_tanh(qv[i].x + kv.x);
      part += wv[i].y * fast_tanh(qv[i].y + kv.y);
      part += wv[i].z * fast_tanh(qv[i].z + kv.z);
      part += wv[i].w * fast_tanh(qv[i].w + kv.w);
    }
#pragma unroll
    for (int off = 16; off > 0; off >>= 1)
      part += __shfl_xor(part, off, 32);
    if (lane == 0) sc[s] = part + bias;   // TEMPERATURE == 1
  }
  __syncthreads();

  // --- softmax over sc[0..1023] ---
  float m = -3.402823466e38f;
  for (int s = tid; s < Sc; s += 256) m = fmaxf(m, sc[s]);
  red[tid] = m;
  __syncthreads();
  for (int w = 128; w > 0; w >>= 1) {
    if (tid < w) red[tid] = fmaxf(red[tid], red[tid + w]);
    __syncthreads();
  }
  const float mx = red[0];
  __syncthreads();

  float ssum = 0.0f;
  for (int s = tid; s < Sc; s += 256) {
    const float e = __expf(sc[s] - mx);
    sc[s] = e;
    ssum += e;
  }
  red[tid] = ssum;
  __syncthreads();
  for (int w = 128; w > 0; w >>= 1) {
    if (tid < w) red[tid] += red[tid + w];
    __syncthreads();
  }
  const float inv = 1.0f / red[0];

  float* wrow = wout + (size_t)bq * Sc;
  for (int s = tid; s < Sc; s += 256) wrow[s] = sc[s] * inv;
}

// ---------------------------------------------------------------------------
extern "C" void kernel_launch(void* const* d_in, const int* in_sizes, int n_in,
                              void* d_out, int out_size, void* d_ws, size_t ws_size,
                              hipStream_t stream) {
  const float* query = (const float*)d_in[0];  // [B,Q,H]
  const float* key_  = (const float*)d_in[1];  // [B,S,H]
  const float* value = (const float*)d_in[2];  // [B,S,H]
  const float* Wq    = (const float*)d_in[3];  // [H,H]
  const float* bq    = (const float*)d_in[4];  // [H]
  const float* Wk    = (const float*)d_in[5];  // [H,H]
  const float* bk    = (const float*)d_in[6];  // [H]
  const float* we    = (const float*)d_in[7];  // [H]
  const float* be    = (const float*)d_in[8];  // scalar

  float* ctx = (float*)d_out;                       // [B,Q,H]
  float* wts = ctx + (size_t)Bc * Qc * Hc;          // [B,Q,S]

  float* qp = (float*)d_ws;                         // [B*Q, H]
  float* kp = qp + (size_t)Bc * Qc * Hc;            // [B*S, H]

  const dim3 blk(128);

  // qp = query @ Wq + bq   : M=256,  N=512, K=512
  wmma_gemm_bias<<<dim3(Hc / 64, (Bc * Qc) / 64, 1), blk, 0, stream>>>(
      query, Wq, bq, qp, Bc * Qc, Hc, Hc, 0, 0, 0);

  // kp = key @ Wk + bk     : M=4096, N=512, K=512
  wmma_gemm_bias<<<dim3(Hc / 64, (Bc * Sc) / 64, 1), blk, 0, stream>>>(
      key_, Wk, bk, kp, Bc * Sc, Hc, Hc, 0, 0, 0);

  // scores + softmax -> weights (written straight into output slab)
  attn_scores_softmax<<<dim3(Bc * Qc), dim3(256), 0, stream>>>(qp, kp, we, be, wts);

  // context[b] = weights[b] @ value[b] : M=64, N=512, K=1024, batched over b
  wmma_gemm_bias<<<dim3(Hc / 64, 1, Bc), blk, 0, stream>>>(
      wts, value, nullptr, ctx, Qc, Hc, Sc,
      (long long)Qc * Sc, (long long)Sc * Hc, (long long)Qc * Hc);
}